// ScipyConv2d_67001489817769
// MI455X (gfx1250) — compile-verified
//
#include <hip/hip_runtime.h>
#include <hip/hip_bf16.h>
#include <math.h>

#define PH      2
#define N_ROIS  2048
#define H_IMG   258
#define W_IMG   8194
#define HF      256      // H_IMG - 2
#define WF      8192     // W_IMG - 2

#define TILES_PER_WAVE 16   // persistent wave: 16 column tiles per wave

typedef __attribute__((ext_vector_type(2))) float v2f;
typedef __attribute__((ext_vector_type(8))) float v8f;

// -----------------------------------------------------------------------------
// Stage 1: 3x3 VALID conv as implicit-Toeplitz GEMM on the f32 WMMA pipe.
//
//   out[h,w] = sum_{i=0..2} sum_{w'=0..17} img[h0+h+i][w0+w'] * T_i[w' - w]
// with T_i[j] = k[i][j] for j in [0,3), else 0.  K = (i, w') flattened, each
// i-segment padded to 20 -> 15 chained V_WMMA_F32_16X16X4_F32 per 16x16 tile.
//
// Persistent-wave version: the banded-Toeplitz B operands depend only on
// (lane, i, chunk), so the 15 B pairs are built ONCE per wave (cndmask
// chains), then the wave sweeps 16 consecutive column tiles reusing them.
// Steady-state loop body: 12 global_load_b64 + 6 scalar loads (A) + 15 v_wmma + 8 stores.
//
// Layouts (ISA 7.12.2, wave32):
//   A 16x4 f32 : lane m=lane&15 is row M; lanes 0-15 hold K={0,1},
//                lanes 16-31 hold K={2,3} (VGPR pair).
//   B 4x16 f32 : lane n=lane&15 is col N; half-waves mirror A's K split.
//   C 16x16 f32: VGPR r -> M=r (lanes 0-15) / M=r+8 (lanes 16-31), N=lane&15.
// -----------------------------------------------------------------------------
__global__ __launch_bounds__(256) void conv3x3_wmma_kernel(
    const float* __restrict__ img,
    const float* __restrict__ kw,      // 9 filter taps, row-major [3][3]
    const float* __restrict__ kb,      // 1 bias
    float* __restrict__ feat)          // [HF][WF]
{
    const int lane  = threadIdx.x & 31;
    const int wave  = blockIdx.x * 8 + (threadIdx.x >> 5);  // 512 waves
    const int tileh = wave >> 5;                            // 16 row-tiles
    const int grp   = wave & 31;                            // 32 col groups
    const int h0    = tileh * 16;

    const int m     = lane & 15;   // A row / B col / C col within tile
    const int khalf = lane >> 4;   // which K pair this half-wave holds

    // Uniform filter taps + bias (scalar loads).
    float t[3][3];
    #pragma unroll
    for (int i = 0; i < 3; ++i)
        #pragma unroll
        for (int j = 0; j < 3; ++j)
            t[i][j] = kw[i * 3 + j];
    const float bias = kb[0];

    // Hoisted Toeplitz B operands: tap j = w' - n, nonzero for j in [0,3).
    // Built once per wave; 15 v2f pairs live in registers across the sweep.
    v2f B[3][5];
    #pragma unroll
    for (int i = 0; i < 3; ++i) {
        #pragma unroll
        for (int ch = 0; ch < 5; ++ch) {
            const int wp = ch * 4 + 2 * khalf;
            const int j0 = wp - m;
            const int j1 = wp + 1 - m;
            B[i][ch].x = (j0 == 0) ? t[i][0] : (j0 == 1) ? t[i][1]
                       : (j0 == 2) ? t[i][2] : 0.0f;
            B[i][ch].y = (j1 == 0) ? t[i][0] : (j1 == 1) ? t[i][1]
                       : (j1 == 2) ? t[i][2] : 0.0f;
        }
    }

    // Sweep 16 consecutive column tiles (uniform trip count: EXEC stays all-1s).
    #pragma unroll 1
    for (int tw = 0; tw < TILES_PER_WAVE; ++tw) {
        const int w0 = (grp * TILES_PER_WAVE + tw) * 16;

        v8f c = {};                    // f32 16x16 accumulator (8 VGPRs)
        #pragma unroll
        for (int i = 0; i < 3; ++i) {
            const long rowbase = (long)(h0 + m + i) * W_IMG;
            #pragma unroll
            for (int ch = 0; ch < 5; ++ch) {
                const int wp = ch * 4 + 2 * khalf;   // w' of K pair element 0
                v2f a;
                if (ch < 4) {
                    // 8B-aligned pair: img[h0+m+i][w0+wp .. w0+wp+1]
                    a = *(const v2f*)(img + rowbase + (w0 + wp));
                } else {
                    // last chunk may touch w0+18/19 -> clamp (B coef is 0 there)
                    int c0 = w0 + wp;     c0 = c0 < (W_IMG - 1) ? c0 : (W_IMG - 1);
                    int c1 = w0 + wp + 1; c1 = c1 < (W_IMG - 1) ? c1 : (W_IMG - 1);
                    a.x = img[rowbase + c0];
                    a.y = img[rowbase + c1];
                }
                // D = A*B + C (f32, RNE) — chained accumulation on matrix pipe
                c = __builtin_amdgcn_wmma_f32_16x16x4_f32(
                        false, a, false, B[i][ch], (short)0, c, false, false);
            }
        }

        // Store C tile (+bias): VGPR r -> row h0 + r + 8*khalf, col w0 + m.
        const int ccol = w0 + m;
        #pragma unroll
        for (int r = 0; r < 8; ++r) {
            const int crow = h0 + r + 8 * khalf;
            feat[(long)crow * WF + ccol] = c[r] + bias;
        }
    }
}

// -----------------------------------------------------------------------------
// Stage 2: ROI max pool.  One wave per (roi, ph, pw) bin.  Geometry replicates
// the host-side float64 numpy math bit-for-bit (rint == np.round half-to-even).
// -----------------------------------------------------------------------------
__global__ __launch_bounds__(256) void roi_maxpool_kernel(
    const float* __restrict__ rois,   // [N_ROIS][5]
    const float* __restrict__ feat,   // [HF][WF]
    float* __restrict__ out,          // [N_ROIS][1][PH][mpw]
    int mpw, int nbins)
{
    const int lane = threadIdx.x & 31;
    const int bin  = blockIdx.x * 8 + (threadIdx.x >> 5);
    if (bin >= nbins) return;

    const int pw = bin % mpw;
    const int ph = (bin / mpw) % PH;
    const int n  = bin / (mpw * PH);

    // Geometry in f64 (wave-uniform; every lane computes it).
    const double x1 = (double)rois[n * 5 + 1] * 0.25;
    const double y1 = (double)rois[n * 5 + 2] * 0.0625;
    const double x2 = (double)rois[n * 5 + 3] * 0.25;
    const double y2 = (double)rois[n * 5 + 4] * 0.0625;

    const double rsw = rint(x1);           // np.round -> round-half-to-even
    const double rsh = rint(y1);
    const double rw  = fmax(x2 - x1, 1.0);
    const double rh  = fmax(y2 - y1, 1.0);
    long pwn = (long)ceil(rw / rh * (double)PH);
    if (pwn < 1) pwn = 1;
    const double bsw = rw / (double)pwn;
    const double bsh = rh / (double)PH;
    const bool skip = (ceil(2.0 * bsh) - floor(bsh) == 0.0) ||
                      (ceil(2.0 * bsw) - floor(bsw) == 0.0);

    const int hs = (int)fmin(fmax(floor((double)ph * bsh) + rsh, 0.0), (double)HF);
    const int he = (int)fmin(fmax(ceil((double)(ph + 1) * bsh) + rsh, 0.0), (double)HF);
    const int ws = (int)fmin(fmax(floor((double)pw * bsw) + rsw, 0.0), (double)WF);
    const int we = (int)fmin(fmax(ceil((double)(pw + 1) * bsw) + rsw, 0.0), (double)WF);

    const bool valid = (pw < pwn) && !skip && (he > hs) && (we > ws);

    float r = -__builtin_inff();
    if (valid) {
        for (int h = hs; h < he; ++h) {
            const float* rowp = feat + (long)h * WF;
            if (h + 1 < he)   // gfx1250 global_prefetch of next row's span
                __builtin_prefetch(feat + (long)(h + 1) * WF + ws + lane, 0, 1);
            for (int w = ws + lane; w < we; w += 32)
                r = fmaxf(r, rowp[w]);
        }
    }
    // wave32 max-reduction
    #pragma unroll
    for (int off = 16; off >= 1; off >>= 1)
        r = fmaxf(r, __shfl_xor(r, off, 32));

    if (lane == 0)
        out[bin] = valid ? r : 0.0f;   // bin index == n*PH*mpw + ph*mpw + pw
}

// -----------------------------------------------------------------------------
// inputs: d_in[0]=img (1*1*258*8194 f32), d_in[1]=rois (2048*5 f32),
//         d_in[2]=conv_w (9 f32), d_in[3]=conv_b (1 f32)
// d_ws: feature map scratch, 256*8192*4 = 8 MB.
// -----------------------------------------------------------------------------
extern "C" void kernel_launch(void* const* d_in, const int* in_sizes, int n_in,
                              void* d_out, int out_size, void* d_ws, size_t ws_size,
                              hipStream_t stream) {
    const float* img  = (const float*)d_in[0];
    const float* rois = (const float*)d_in[1];
    const float* kw   = (const float*)d_in[2];
    const float* kb   = (const float*)d_in[3];
    float* out  = (float*)d_out;
    float* feat = (float*)d_ws;                 // 8 MB << ws_size

    // 8192 tiles of 16x16; persistent waves sweep 16 tiles each:
    // 512 waves = 64 blocks * 8 waves.
    conv3x3_wmma_kernel<<<64, 256, 0, stream>>>(img, kw, kb, feat);

    const int mpw   = out_size / (N_ROIS * PH); // recover mpw from output size
    const int nbins = N_ROIS * PH * mpw;
    const int blocks = (nbins + 7) / 8;         // 8 waves per block
    roi_maxpool_kernel<<<blocks, 256, 0, stream>>>(rois, feat, out, mpw, nbins);
}